// HadamardTransform_46540265619692
// MI455X (gfx1250) — compile-verified
//
#include <hip/hip_runtime.h>

// FWHT over HEAD_DIM=128 blocks, fp32, scaled by 1/sqrt(128).
// H128 = H16 (x) H8.  Per wave: two heads -> one 16x16 tile.
//   Stage 1: W = (s*H16) * [V_A | V_B]   via 4x v_wmma_f32_16x16x4_f32 (fp32-exact)
//   Stage 2: H8 along j (lane bits 0..2) via ds_swizzle_b32 xor-butterflies.
// Memory-bound: 512 MiB traffic -> ~22 us floor at 23.3 TB/s.

typedef float v2f __attribute__((ext_vector_type(2)));
typedef float v8f __attribute__((ext_vector_type(8)));

// ds_swizzle group-of-32 encoding: offset = xor<<10 | or<<5 | and ; and=0x1f
#define SWZ_XOR1 0x041f
#define SWZ_XOR2 0x081f
#define SWZ_XOR4 0x101f

template <int IMM, int BIT>
__device__ __forceinline__ void bfly8(float (&r)[8], int lane) {
    const bool hi = (lane & BIT) != 0;
#pragma unroll
    for (int i = 0; i < 8; ++i) {
        float t = __int_as_float(
            __builtin_amdgcn_ds_swizzle(__float_as_int(r[i]), IMM));
        // partner-with-bit-clear gets a+b, bit-set gets a-b (Sylvester order)
        r[i] = hi ? (t - r[i]) : (r[i] + t);
    }
}

__global__ void __launch_bounds__(256)
fwht128_wmma_kernel(const float* __restrict__ in, float* __restrict__ out,
                    int ntiles) {
    const int lane = threadIdx.x & 31;
    const int wave = threadIdx.x >> 5;
    const int tile = blockIdx.x * (blockDim.x >> 5) + wave;  // wave-uniform
    if (tile >= ntiles) return;  // whole wave exits together; EXEC stays all-1

    const float* __restrict__ base  = in  + (size_t)tile * 256;  // 2 heads
    float* __restrict__       obase = out + (size_t)tile * 256;

    const int col  = lane & 15;   // A-row (M) and B-column (N) for this lane
    const int half = lane >> 4;   // lane 16..31 holds K+2 / M+8 halves
    const float s  = 0.08838834764831845f;  // 1/sqrt(128), folded into H16

    // A-operand: (s*H16)[row, k], chunked K=4.  H16[i,k] = (-1)^popc(i&k).
    // Layout: lanes 0-15 -> M=lane, {v0,v1} = K = 4c+{0,1};
    //         lanes 16-31 -> M=lane-16, K = 4c+{2,3}.
    v2f A[4];
#pragma unroll
    for (int c = 0; c < 4; ++c) {
        const int k0 = 4 * c + 2 * half;
        A[c][0] = (__builtin_popcount(col & k0) & 1) ? -s : s;
        A[c][1] = (__builtin_popcount(col & (k0 + 1)) & 1) ? -s : s;
    }

    // B-operand loaded straight from HBM in WMMA register layout.
    // X[k, n] = (n<8) ? headA[k*8+n] : headB[k*8+n-8];  headB = headA+128.
    const int coloff = ((col & 8) << 4) + (col & 7);

    v8f acc = {};
#pragma unroll
    for (int c = 0; c < 4; ++c) {
        const int kx = 4 * c + 2 * half;
        v2f b;
        b[0] = base[coloff + kx * 8];
        b[1] = base[coloff + (kx + 1) * 8];
        // (neg_a, A, neg_b, B, c_mod, C, reuse_a, reuse_b)
        acc = __builtin_amdgcn_wmma_f32_16x16x4_f32(
            false, A[c], false, b, (short)0, acc, false, false);
    }

    // acc layout: VGPR i, lanes 0-15 -> W[i, col]; lanes 16-31 -> W[i+8, col].
    float r[8];
#pragma unroll
    for (int i = 0; i < 8; ++i) r[i] = acc[i];

    // H8 along j = lane bits 0..2 (stays inside each head's 8-lane group).
    bfly8<SWZ_XOR1, 1>(r, lane);
    bfly8<SWZ_XOR2, 2>(r, lane);
    bfly8<SWZ_XOR4, 4>(r, lane);

    // Y[m, j] -> head element m*8 + j
#pragma unroll
    for (int i = 0; i < 8; ++i) {
        const int m = i + 8 * half;
        obase[coloff + m * 8] = r[i];
    }
}

extern "C" void kernel_launch(void* const* d_in, const int* in_sizes, int n_in,
                              void* d_out, int out_size, void* d_ws,
                              size_t ws_size, hipStream_t stream) {
    const float* in = (const float*)d_in[0];
    float* out = (float*)d_out;
    const int n = in_sizes[0];          // 16384*4096
    const int ntiles = n / 256;         // 2 heads (256 floats) per wave
    const int waves_per_block = 8;      // 256 threads = 8 wave32
    dim3 block(32 * waves_per_block);
    dim3 grid((ntiles + waves_per_block - 1) / waves_per_block);
    hipLaunchKernelGGL(fwht128_wmma_kernel, grid, block, 0, stream, in, out,
                       ntiles);
}